// Ipfl_26482768347622
// MI455X (gfx1250) — compile-verified
//
#include <hip/hip_runtime.h>
#include <math.h>

// ---------------------------------------------------------------------------
// CDNA5 / gfx1250 implementation.
// Heavy work: two FP32 GEMMs via V_WMMA_F32_16X16X4_F32 with fused
// L2-distance epilogue; b128 loads feed two WMMA k-steps each, software-
// pipelined one iteration ahead so load latency hides behind the WMMA burst.
// Light work: top-3 / first-trusted-of-top-15 selection + deterministic mean.
// ---------------------------------------------------------------------------

typedef __attribute__((ext_vector_type(2))) float v2f;
typedef __attribute__((ext_vector_type(4))) float v4f;
typedef __attribute__((ext_vector_type(8))) float v8f;

#define WMMA_F32_16x16x4(A, B, C) \
    __builtin_amdgcn_wmma_f32_16x16x4_f32(false, (A), false, (B), 0, (C), false, false)

__device__ __forceinline__ v2f lo2(v4f x) { return __builtin_shufflevector(x, x, 0, 1); }
__device__ __forceinline__ v2f hi2(v4f x) { return __builtin_shufflevector(x, x, 2, 3); }

// ---------------------------------------------------------------------------
// Row sum-of-squares: one 256-thread block per row.
// ---------------------------------------------------------------------------
__global__ void rownorm_kernel(const float* __restrict__ X,
                               float* __restrict__ out, int K) {
    __shared__ float red[256];
    const int row = blockIdx.x;
    const float* p = X + (size_t)row * K;
    float s = 0.0f;
    for (int k = threadIdx.x; k < K; k += blockDim.x) {
        float v = p[k];
        s += v * v;
    }
    red[threadIdx.x] = s;
    __syncthreads();
    for (int off = blockDim.x >> 1; off > 0; off >>= 1) {
        if ((int)threadIdx.x < off) red[threadIdx.x] += red[threadIdx.x + off];
        __syncthreads();
    }
    if (threadIdx.x == 0) out[row] = red[0];
}

// ---------------------------------------------------------------------------
// G = A(MxK) * Bm(NxK)^T via FP32 WMMA; fused epilogue:
//   D[m][n] = sqrt(max(a2[m] + b2[n] - 2*G[m][n], 0))
// One wave per 32x32 output tile; 2x2 register subtiles.
//
// K-permutation trick: per k-base (step 8), lane-group g = lane>>4 loads a
// v4f at row + k + 4*g (global_load_b128). The GEMM reduction is invariant
// under permuting global-k -> WMMA k-slot as long as A and B agree, so:
//   WMMA #0 consumes v4f elements {0,1}  (global k: {k,k+1} | {k+4,k+5})
//   WMMA #1 consumes v4f elements {2,3}  (global k: {k+2,k+3} | {k+6,k+7})
// -> 4 b128 loads feed 8 WMMAs per iteration, prefetched one iteration ahead.
// ---------------------------------------------------------------------------
__global__ void __launch_bounds__(32)
gemm_dist_kernel(const float* __restrict__ A, const float* __restrict__ Bm,
                 const float* __restrict__ a2, const float* __restrict__ b2,
                 float* __restrict__ Dout, int N, int K) {
    const int i0   = blockIdx.y * 32;
    const int j0   = blockIdx.x * 32;
    const int lane = threadIdx.x;
    const int r    = lane & 15;
    const int g4   = (lane >> 4) << 2;  // 0 or 4

    const float* Ar0 = A  + (size_t)(i0 + r)      * K + g4;
    const float* Ar1 = A  + (size_t)(i0 + 16 + r) * K + g4;
    const float* Br0 = Bm + (size_t)(j0 + r)      * K + g4;
    const float* Br1 = Bm + (size_t)(j0 + 16 + r) * K + g4;

    v8f acc00 = {}, acc01 = {}, acc10 = {}, acc11 = {};

    // Prologue: fragments for k = 0.
    v4f qa0 = *(const v4f*)(Ar0);
    v4f qa1 = *(const v4f*)(Ar1);
    v4f qb0 = *(const v4f*)(Br0);
    v4f qb1 = *(const v4f*)(Br1);

#pragma unroll 2
    for (int k = 8; k < K; k += 8) {
        // Prefetch next iteration's fragments first; the 8 WMMAs below only
        // depend on the already-resident q* registers, so these loads retire
        // under the WMMA burst instead of stalling mid-iteration.
        v4f na0 = *(const v4f*)(Ar0 + k);
        v4f na1 = *(const v4f*)(Ar1 + k);
        v4f nb0 = *(const v4f*)(Br0 + k);
        v4f nb1 = *(const v4f*)(Br1 + k);

        v2f a0 = lo2(qa0), a1 = lo2(qa1), b0 = lo2(qb0), b1 = lo2(qb1);
        acc00 = WMMA_F32_16x16x4(a0, b0, acc00);
        acc01 = WMMA_F32_16x16x4(a0, b1, acc01);
        acc10 = WMMA_F32_16x16x4(a1, b0, acc10);
        acc11 = WMMA_F32_16x16x4(a1, b1, acc11);

        a0 = hi2(qa0); a1 = hi2(qa1); b0 = hi2(qb0); b1 = hi2(qb1);
        acc00 = WMMA_F32_16x16x4(a0, b0, acc00);
        acc01 = WMMA_F32_16x16x4(a0, b1, acc01);
        acc10 = WMMA_F32_16x16x4(a1, b0, acc10);
        acc11 = WMMA_F32_16x16x4(a1, b1, acc11);

        qa0 = na0; qa1 = na1; qb0 = nb0; qb1 = nb1;
    }

    // Epilogue iteration: consume the last prefetched fragments.
    {
        v2f a0 = lo2(qa0), a1 = lo2(qa1), b0 = lo2(qb0), b1 = lo2(qb1);
        acc00 = WMMA_F32_16x16x4(a0, b0, acc00);
        acc01 = WMMA_F32_16x16x4(a0, b1, acc01);
        acc10 = WMMA_F32_16x16x4(a1, b0, acc10);
        acc11 = WMMA_F32_16x16x4(a1, b1, acc11);

        a0 = hi2(qa0); a1 = hi2(qa1); b0 = hi2(qb0); b1 = hi2(qb1);
        acc00 = WMMA_F32_16x16x4(a0, b0, acc00);
        acc01 = WMMA_F32_16x16x4(a0, b1, acc01);
        acc10 = WMMA_F32_16x16x4(a1, b0, acc10);
        acc11 = WMMA_F32_16x16x4(a1, b1, acc11);
    }

    // C/D layout: VGPR rr -> M = rr + 8*(lane>=16), N = lane&15.
    const int n  = lane & 15;
    const int mh = (lane >> 4) << 3;  // 0 or 8
    const float bn0 = b2[j0 + n];
    const float bn1 = b2[j0 + 16 + n];
#pragma unroll
    for (int rr = 0; rr < 8; ++rr) {
        const int m0 = i0 + mh + rr;
        const int m1 = i0 + 16 + mh + rr;
        const float am0 = a2[m0];
        const float am1 = a2[m1];
        float gacc;
        gacc = acc00[rr];
        Dout[(size_t)m0 * N + j0 + n]      = sqrtf(fmaxf(am0 + bn0 - 2.0f * gacc, 0.0f));
        gacc = acc01[rr];
        Dout[(size_t)m0 * N + j0 + 16 + n] = sqrtf(fmaxf(am0 + bn1 - 2.0f * gacc, 0.0f));
        gacc = acc10[rr];
        Dout[(size_t)m1 * N + j0 + n]      = sqrtf(fmaxf(am1 + bn0 - 2.0f * gacc, 0.0f));
        gacc = acc11[rr];
        Dout[(size_t)m1 * N + j0 + 16 + n] = sqrtf(fmaxf(am1 + bn1 - 2.0f * gacc, 0.0f));
    }
}

// ---------------------------------------------------------------------------
// For each center: indices of 3 smallest distances (stable tie order: first
// index wins, matching jnp.argsort). One thread per center row.
// ---------------------------------------------------------------------------
__global__ void near3_kernel(const float* __restrict__ Dcc,
                             int* __restrict__ nearOut, int C) {
    const int c = blockIdx.x * blockDim.x + threadIdx.x;
    if (c >= C) return;
    const float* row = Dcc + (size_t)c * C;
    float v0 = 3.4e38f, v1 = 3.4e38f, v2 = 3.4e38f;
    int   i0 = -1, i1 = -1, i2 = -1;
    for (int j = 0; j < C; ++j) {
        float v = row[j];
        if (v < v0)      { v2 = v1; i2 = i1; v1 = v0; i1 = i0; v0 = v; i0 = j; }
        else if (v < v1) { v2 = v1; i2 = i1; v1 = v;  i1 = j; }
        else if (v < v2) { v2 = v;  i2 = j; }
    }
    nearOut[3 * c + 0] = i0;
    nearOut[3 * c + 1] = i1;
    nearOut[3 * c + 2] = i2;
}

// ---------------------------------------------------------------------------
// One wave per feature row b (label = b/NUM, NUM=2):
//   same = D_fc[b, label]
//   walk the 15 nearest non-label centers in ascending (value, index) order
//   (== stable argsort order); first trusted one gives min_diff, else 0.
//   hinge = max(1 + same - min_diff, 0) written per row (deterministic).
// ---------------------------------------------------------------------------
__global__ void __launch_bounds__(32)
hinge_kernel(const float* __restrict__ Dfc, const int* __restrict__ near3v,
             float* __restrict__ hvals, int C) {
    const int b     = blockIdx.x;
    const int lane  = threadIdx.x;
    const int label = b >> 1;  // NUM = 2
    const float* row = Dfc + (size_t)b * C;
    const float same = row[label];

    float last_v = -1.0f;
    int   last_c = -1;
    float min_diff = 0.0f;
    bool  found = false;

    for (int it = 0; it < 15 && !found; ++it) {
        // Per-lane min over its strided slice, restricted to (v,c) strictly
        // greater (lexicographically) than the previously selected entry.
        float v = 3.4e38f;
        int   c = 0x7fffffff;
        for (int j = lane; j < C; j += 32) {
            if (j == label) continue;
            float d = row[j];
            bool gt = (d > last_v) || (d == last_v && j > last_c);
            if (gt && (d < v || (d == v && j < c))) { v = d; c = j; }
        }
        // Wave32 lexicographic argmin reduction.
        for (int off = 16; off > 0; off >>= 1) {
            float ov = __shfl_xor(v, off, 32);
            int   oc = __shfl_xor(c, off, 32);
            if (ov < v || (ov == v && oc < c)) { v = ov; c = oc; }
        }
        if (c == 0x7fffffff) break;  // no candidates left
        bool trusted = !(near3v[3 * c + 0] == label ||
                         near3v[3 * c + 1] == label ||
                         near3v[3 * c + 2] == label);
        if (trusted) { min_diff = v; found = true; }
        else         { last_v = v; last_c = c; }
    }

    if (lane == 0) hvals[b] = fmaxf(1.0f + same - min_diff, 0.0f);
}

// ---------------------------------------------------------------------------
// Deterministic fixed-order mean reduction (single block).
// ---------------------------------------------------------------------------
__global__ void reduce_mean_kernel(const float* __restrict__ h,
                                   float* __restrict__ out, int B) {
    __shared__ float red[256];
    float s = 0.0f;
    for (int i = threadIdx.x; i < B; i += 256) s += h[i];
    red[threadIdx.x] = s;
    __syncthreads();
    for (int off = 128; off > 0; off >>= 1) {
        if ((int)threadIdx.x < off) red[threadIdx.x] += red[threadIdx.x + off];
        __syncthreads();
    }
    if (threadIdx.x == 0) *out = red[0] * (1.0f / (float)B);
}

// ---------------------------------------------------------------------------
extern "C" void kernel_launch(void* const* d_in, const int* in_sizes, int n_in,
                              void* d_out, int out_size, void* d_ws, size_t ws_size,
                              hipStream_t stream) {
    (void)in_sizes; (void)n_in; (void)out_size; (void)ws_size;

    const float* feature = (const float*)d_in[0];  // 4096 x 2048
    const float* centers = (const float*)d_in[1];  // 2048 x 2048
    float* out = (float*)d_out;

    const int B = 4096, C = 2048, D = 2048;

    // Workspace layout (~48 MB):
    float* a2      = (float*)d_ws;                 // B
    float* b2      = a2 + B;                       // C
    float* Dfc     = b2 + C;                       // B*C
    float* Dcc     = Dfc + (size_t)B * C;          // C*C
    int*   nearbuf = (int*)(Dcc + (size_t)C * C);  // 3*C
    float* hvals   = (float*)(nearbuf + 3 * C);    // B

    rownorm_kernel<<<B, 256, 0, stream>>>(feature, a2, D);
    rownorm_kernel<<<C, 256, 0, stream>>>(centers, b2, D);

    gemm_dist_kernel<<<dim3(C / 32, B / 32), 32, 0, stream>>>(
        feature, centers, a2, b2, Dfc, C, D);
    gemm_dist_kernel<<<dim3(C / 32, C / 32), 32, 0, stream>>>(
        centers, centers, b2, b2, Dcc, C, D);

    near3_kernel<<<(C + 255) / 256, 256, 0, stream>>>(Dcc, nearbuf, C);
    hinge_kernel<<<B, 32, 0, stream>>>(Dfc, nearbuf, hvals, C);
    reduce_mean_kernel<<<1, 256, 0, stream>>>(hvals, out, B);
}